// GraphVAE_3315714752918
// MI455X (gfx1250) — compile-verified
//
#include <hip/hip_runtime.h>
#include <math.h>

#define NN 2048
#define EE 65536
#define FD 7
#define SD 32
#define LD 16
#define RR 3

typedef __attribute__((ext_vector_type(2))) float v2f;
typedef __attribute__((ext_vector_type(8))) float v8f;

__device__ __forceinline__ float sigmoidf_(float x) { return 1.0f / (1.0f + __expf(-x)); }

// async copy of one 16B chunk from the global address saddr + vaddr to the LDS address in ldsoff
__device__ __forceinline__ void async_b128(unsigned ldsoff, unsigned vaddr, const void* saddr) {
  asm volatile("global_load_async_to_lds_b128 %0, %1, %2"
               :: "v"(ldsoff), "v"(vaddr), "s"(saddr) : "memory");
}
__device__ __forceinline__ void wait_async0() {
  asm volatile("s_wait_asynccnt 0x0" ::: "memory");
}

// ---------------- encoder: state = relu(x @ W_in + b_in) ----------------
__global__ void enc_kernel(const float* __restrict__ x, const float* __restrict__ W,
                           const float* __restrict__ b, float* __restrict__ state) {
  int idx = blockIdx.x * blockDim.x + threadIdx.x;   // n*SD + s
  if (idx >= NN * SD) return;
  int n = idx >> 5, s = idx & 31;
  float acc = b[s];
#pragma unroll
  for (int f = 0; f < FD; ++f) acc += x[n * FD + f] * W[f * SD + s];
  state[idx] = fmaxf(acc, 0.0f);
}

// ---------------- message = relu(state @ Wm + bm); also zero agg ----------------
// one wave per 16-node group; 2 output tiles of 16; K=32 via 8x WMMA_F32_16X16X4_F32
// Wm (32x32 = 4KB) staged once per block into LDS via async-to-LDS.
__global__ void msg_kernel(const float* __restrict__ state, const float* __restrict__ Wm,
                           const float* __restrict__ bm, float* __restrict__ message,
                           float* __restrict__ agg) {
  __shared__ __align__(16) float Wl[SD * SD];
  unsigned lds0 = (unsigned)(size_t)Wl;
#pragma unroll
  for (int i = 0; i < 2; ++i) {                     // 256 chunks of 16B, 128 threads
    unsigned idx = threadIdx.x + i * 128;
    async_b128(lds0 + idx * 16, idx * 16, Wm);
  }
  wait_async0();
  __syncthreads();

  int lane = threadIdx.x & 31;
  int wave = threadIdx.x >> 5;
  int base = (blockIdx.x * (blockDim.x >> 5) + wave) * 16;
  int ln = lane & 15, hi = lane >> 4, kb = hi << 1;

  v2f a[8];
#pragma unroll
  for (int ks = 0; ks < 8; ++ks)
    a[ks] = *(const v2f*)(state + (base + ln) * SD + ks * 4 + kb);

#pragma unroll
  for (int nt = 0; nt < 2; ++nt) {
    int n = nt * 16 + ln;
    v8f c = {};
#pragma unroll
    for (int ks = 0; ks < 8; ++ks) {
      int k = ks * 4 + kb;
      v2f bfrag;
      bfrag.x = Wl[k * SD + n];
      bfrag.y = Wl[(k + 1) * SD + n];
      c = __builtin_amdgcn_wmma_f32_16x16x4_f32(false, a[ks], false, bfrag,
                                                (short)0, c, false, false);
    }
    float bias = bm[n];
#pragma unroll
    for (int v = 0; v < 8; ++v) {
      int row = base + v + hi * 8;
      message[row * SD + n] = fmaxf(c[v] + bias, 0.0f);
      agg[row * SD + n] = 0.0f;
    }
  }
}

// ---------------- edge scatter: agg[dst] += message[src] ----------------
__global__ void scatter_kernel(const int* __restrict__ ei, const float* __restrict__ message,
                               float* __restrict__ agg) {
  unsigned idx = blockIdx.x * blockDim.x + threadIdx.x;
  if (idx >= (unsigned)EE * SD) return;
  int e = idx >> 5, k = idx & 31;
  int s = ei[e];
  int d = ei[EE + e];
  atomicAdd(&agg[d * SD + k], message[s * SD + k]);
}

// ---------------- GRU: gx = agg@Wih, gh = state@Whh, gates, residual ----------------
// Wih+Whh (2 x 32x96 = 24KB) staged once per block into LDS via async-to-LDS.
__global__ void gru_kernel(const float* __restrict__ agg, const float* __restrict__ Wih,
                           const float* __restrict__ Whh, const float* __restrict__ bih,
                           const float* __restrict__ bhh, float* __restrict__ state) {
  __shared__ __align__(16) float Wl[2 * SD * 96];
  float* WihL = Wl;
  float* WhhL = Wl + SD * 96;
  unsigned lds0 = (unsigned)(size_t)Wl;
#pragma unroll
  for (int i = 0; i < 6; ++i) {                     // 768 chunks per matrix, 128 threads
    unsigned idx = threadIdx.x + i * 128;
    async_b128(lds0 + idx * 16, idx * 16, Wih);
    async_b128(lds0 + 12288 + idx * 16, idx * 16, Whh);
  }
  wait_async0();
  __syncthreads();

  int lane = threadIdx.x & 31;
  int wave = threadIdx.x >> 5;
  int base = (blockIdx.x * (blockDim.x >> 5) + wave) * 16;
  int ln = lane & 15, hi = lane >> 4, kb = hi << 1;

  v2f aX[8], aH[8];
#pragma unroll
  for (int ks = 0; ks < 8; ++ks) {
    int k = ks * 4 + kb;
    aX[ks] = *(const v2f*)(agg + (base + ln) * SD + k);
    aH[ks] = *(const v2f*)(state + (base + ln) * SD + k);
  }

  v8f gx[6], gh[6];
#pragma unroll
  for (int nt = 0; nt < 6; ++nt) {
    int n = nt * 16 + ln;
    v8f cx = {}, ch = {};
#pragma unroll
    for (int ks = 0; ks < 8; ++ks) {
      int k = ks * 4 + kb;
      v2f bx, bh;
      bx.x = WihL[k * 96 + n];
      bx.y = WihL[(k + 1) * 96 + n];
      bh.x = WhhL[k * 96 + n];
      bh.y = WhhL[(k + 1) * 96 + n];
      cx = __builtin_amdgcn_wmma_f32_16x16x4_f32(false, aX[ks], false, bx,
                                                 (short)0, cx, false, false);
      ch = __builtin_amdgcn_wmma_f32_16x16x4_f32(false, aH[ks], false, bh,
                                                 (short)0, ch, false, false);
    }
    float bbx = bih[n], bbh = bhh[n];
#pragma unroll
    for (int v = 0; v < 8; ++v) { cx[v] += bbx; ch[v] += bbh; }
    gx[nt] = cx;
    gh[nt] = ch;
  }

  // gate math in D-layout: element (row = base+v+hi*8, k = kt*16+ln)
#pragma unroll
  for (int kt = 0; kt < 2; ++kt) {
#pragma unroll
    for (int v = 0; v < 8; ++v) {
      int row = base + v + hi * 8;
      int k = kt * 16 + ln;
      float s = state[row * SD + k];
      float rg = sigmoidf_(gx[kt][v] + gh[kt][v]);
      float zg = sigmoidf_(gx[2 + kt][v] + gh[2 + kt][v]);
      float ng = tanhf(gx[4 + kt][v] + rg * gh[4 + kt][v]);
      float nh = (1.0f - zg) * ng + zg * s;
      state[row * SD + k] = s + nh;
    }
  }
}

// ---------------- heads: mu, logstd ----------------
__global__ void heads_kernel(const float* __restrict__ state, const float* __restrict__ Wmu,
                             const float* __restrict__ bmu, const float* __restrict__ Wls,
                             const float* __restrict__ bls, float* __restrict__ mu,
                             float* __restrict__ ls) {
  int idx = blockIdx.x * blockDim.x + threadIdx.x;   // n*LD + l
  if (idx >= NN * LD) return;
  int n = idx >> 4, l = idx & 15;
  float am = bmu[l], al = bls[l];
#pragma unroll
  for (int k = 0; k < SD; ++k) {
    float s = state[n * SD + k];
    am += s * Wmu[k * LD + l];
    al += s * Wls[k * LD + l];
  }
  mu[idx] = am;
  ls[idx] = al;
}

// ---------------- decoder projections: a = mu@Wd1[:L], b = mu@Wd1[L:] ----------------
__global__ void ab_kernel(const float* __restrict__ mu, const float* __restrict__ Wd1,
                          float* __restrict__ amat, float* __restrict__ bmat) {
  int idx = blockIdx.x * blockDim.x + threadIdx.x;   // n*LD + l
  if (idx >= NN * LD) return;
  int n = idx >> 4, l = idx & 15;
  float aa = 0.0f, bb = 0.0f;
#pragma unroll
  for (int k = 0; k < LD; ++k) {
    float z = mu[n * LD + k];
    aa += z * Wd1[k * LD + l];
    bb += z * Wd1[(LD + k) * LD + l];
  }
  amat[idx] = aa;
  bmat[idx] = bb;
}

// ---------------- decoder: sigmoid(sym(relu(a_i + b_j + bd1) . Wd2 + bd2)) ----------------
// tiles[0]=aI, tiles[1]=bI, tiles[2]=aJ, tiles[3]=bJ, each 16x16 f32, async-staged to LDS.
__global__ void dec_kernel(const float* __restrict__ amat, const float* __restrict__ bmat,
                           const float* __restrict__ bd1, const float* __restrict__ Wd2,
                           const float* __restrict__ bd2, float* __restrict__ out) {
  __shared__ __align__(16) float tiles[4][16][16];
  __shared__ float bc[16], w2[16];
  int tx = threadIdx.x, ty = threadIdx.y;
  int t = ty * 16 + tx;
  int i0 = blockIdx.y * 16, j0 = blockIdx.x * 16;

  unsigned lds0 = (unsigned)(size_t)&tiles[0][0][0];
  if (t < 64) {                                     // 64 chunks of 16B per tile
    unsigned row = (unsigned)t >> 2, quad = (unsigned)t & 3;
    unsigned vaI = ((unsigned)i0 + row) * 64 + quad * 16;   // byte offsets
    unsigned vaJ = ((unsigned)j0 + row) * 64 + quad * 16;
    unsigned ld = (unsigned)t * 16;
    async_b128(lds0 + ld, vaI, amat);                       // aI
    async_b128(lds0 + 1024 + ld, vaI, bmat);                // bI
    async_b128(lds0 + 2048 + ld, vaJ, amat);                // aJ
    async_b128(lds0 + 3072 + ld, vaJ, bmat);                // bJ
  }
  if (ty == 0) { bc[tx] = bd1[tx]; w2[tx] = Wd2[tx]; }
  wait_async0();
  __syncthreads();

  float acc = 0.0f, acc2 = 0.0f;
#pragma unroll
  for (int k = 0; k < 16; ++k) {
    acc  += fmaxf(tiles[0][ty][k] + tiles[3][tx][k] + bc[k], 0.0f) * w2[k];
    acc2 += fmaxf(tiles[2][tx][k] + tiles[1][ty][k] + bc[k], 0.0f) * w2[k];
  }
  out[(i0 + ty) * NN + (j0 + tx)] = sigmoidf_(0.5f * (acc + acc2) + bd2[0]);
}

extern "C" void kernel_launch(void* const* d_in, const int* in_sizes, int n_in,
                              void* d_out, int out_size, void* d_ws, size_t ws_size,
                              hipStream_t stream) {
  const float* x    = (const float*)d_in[0];
  const int*   ei   = (const int*)d_in[1];
  const float* W_in = (const float*)d_in[2];
  const float* b_in = (const float*)d_in[3];
  const float* W_msg = (const float*)d_in[4];
  const float* b_msg = (const float*)d_in[5];
  const float* W_ih = (const float*)d_in[6];
  const float* W_hh = (const float*)d_in[7];
  const float* b_ih = (const float*)d_in[8];
  const float* b_hh = (const float*)d_in[9];
  const float* W_mu = (const float*)d_in[10];
  const float* b_mu = (const float*)d_in[11];
  const float* W_ls = (const float*)d_in[12];
  const float* b_ls = (const float*)d_in[13];
  const float* W_d1 = (const float*)d_in[14];
  const float* b_d1 = (const float*)d_in[15];
  const float* W_d2 = (const float*)d_in[16];
  const float* b_d2 = (const float*)d_in[17];

  float* out = (float*)d_out;
  float* adj = out;
  float* mu  = out + (size_t)NN * NN;
  float* ls  = mu + NN * LD;

  float* ws      = (float*)d_ws;
  float* state   = ws;                 // NN*SD
  float* message = ws + NN * SD;       // NN*SD
  float* agg     = ws + 2 * NN * SD;   // NN*SD
  float* amat    = ws + 3 * NN * SD;   // NN*LD
  float* bmat    = amat + NN * LD;     // NN*LD

  enc_kernel<<<(NN * SD) / 256, 256, 0, stream>>>(x, W_in, b_in, state);

  for (int r = 0; r < RR; ++r) {
    // 2048 nodes / 16 per wave / 4 waves per block = 32 blocks
    msg_kernel<<<32, 128, 0, stream>>>(state, W_msg + r * SD * SD, b_msg + r * SD,
                                       message, agg);
    scatter_kernel<<<(EE * SD) / 256, 256, 0, stream>>>(ei, message, agg);
    gru_kernel<<<32, 128, 0, stream>>>(agg, W_ih + r * SD * 96, W_hh + r * SD * 96,
                                       b_ih + r * 96, b_hh + r * 96, state);
  }

  heads_kernel<<<(NN * LD) / 256, 256, 0, stream>>>(state, W_mu, b_mu, W_ls, b_ls, mu, ls);
  ab_kernel<<<(NN * LD) / 256, 256, 0, stream>>>(mu, W_d1, amat, bmat);
  dec_kernel<<<dim3(NN / 16, NN / 16), dim3(16, 16), 0, stream>>>(amat, bmat, b_d1,
                                                                  W_d2, b_d2, adj);
}